// SomFast_26603027432148
// MI455X (gfx1250) — compile-verified
//
#include <hip/hip_runtime.h>

// ---------------------------------------------------------------------------
// SOM distance + BMU search for MI455X (gfx1250, wave32).
//
//   dist(b,u) = || x_b - w_u + eps ||_2
//             = sqrt( a[b] + wn[u] - 2 * dot(x_b, w_u) )
//   a[b]  = ||x_b||^2 + 2*eps*sum(x_b)
//   wn[u] = ||w_u - eps||^2
//
// The dot() term is a 512x1024x256 f32 GEMM executed with
// V_WMMA_F32_16X16X4_F32 (exact f32 MAC -> argmin matches reference).
// ---------------------------------------------------------------------------

typedef float v2f __attribute__((ext_vector_type(2)));
typedef float v8f __attribute__((ext_vector_type(8)));

#define SOM_EPS   1e-6f
#define SOM_B     512
#define SOM_U     1024   // 32*32 units
#define SOM_S     256
#define SOM_COLS  32

// ---------------------------------------------------------------------------
// Kernel 0: per-row norms. One wave32 per row; each lane handles 8 floats
// (two float4 loads), then a 5-step shfl_xor butterfly reduction.
//   blocks [0,512)      -> a[b]  = ||x||^2 + 2*eps*sum(x)
//   blocks [512,1536)   -> wn[u] = ||w - eps||^2
// ---------------------------------------------------------------------------
__global__ __launch_bounds__(32) void som_norms(const float* __restrict__ x,
                                                const float* __restrict__ w,
                                                float* __restrict__ ws) {
    const int r    = blockIdx.x;
    const int lane = threadIdx.x;

    float sq = 0.0f;   // sum of squares
    float ln = 0.0f;   // linear sum (only used for x rows)

    if (r < SOM_B) {
        const float4* p = (const float4*)(x + (size_t)r * SOM_S) + lane * 2;
        float4 v0 = p[0];
        float4 v1 = p[1];
        sq = v0.x*v0.x + v0.y*v0.y + v0.z*v0.z + v0.w*v0.w
           + v1.x*v1.x + v1.y*v1.y + v1.z*v1.z + v1.w*v1.w;
        ln = v0.x + v0.y + v0.z + v0.w + v1.x + v1.y + v1.z + v1.w;
    } else {
        const float4* p = (const float4*)(w + (size_t)(r - SOM_B) * SOM_S) + lane * 2;
        float4 v0 = p[0];
        float4 v1 = p[1];
        float e = SOM_EPS;
        float a0 = v0.x - e, a1 = v0.y - e, a2 = v0.z - e, a3 = v0.w - e;
        float a4 = v1.x - e, a5 = v1.y - e, a6 = v1.z - e, a7 = v1.w - e;
        sq = a0*a0 + a1*a1 + a2*a2 + a3*a3 + a4*a4 + a5*a5 + a6*a6 + a7*a7;
    }

    // wave32 butterfly reduction
    #pragma unroll
    for (int off = 16; off > 0; off >>= 1) {
        sq += __shfl_xor(sq, off, 32);
        ln += __shfl_xor(ln, off, 32);
    }

    if (lane == 0) {
        if (r < SOM_B) ws[r] = sq + 2.0f * SOM_EPS * ln;
        else           ws[r] = sq;             // ws[512 + u]
    }
}

// ---------------------------------------------------------------------------
// Kernel 1: tiled GEMM + fused distance epilogue.
// 2048 tiles of 16x16 output; one wave32 per tile; 8 waves per block.
// K loop: 64 iterations of V_WMMA_F32_16X16X4_F32.
//
// A-matrix 16x4 f32 lane layout (ISA 7.12.2): lanes 0-15 supply K={k,k+1}
// for row M=lane; lanes 16-31 supply K={k+2,k+3}. B-matrix 4x16 mirrors it
// with N=lane&15. Both fragments are one b64 load per lane per step.
// ---------------------------------------------------------------------------
__global__ __launch_bounds__(256) void som_dist(const float* __restrict__ x,
                                                const float* __restrict__ w,
                                                const float* __restrict__ ws,
                                                float* __restrict__ out) {
    const int lane = threadIdx.x & 31;
    const int wave = threadIdx.x >> 5;
    const int tile = blockIdx.x * 8 + wave;   // 0..2047
    const int tRow = tile >> 6;               // 0..31  (batch tiles)
    const int tCol = tile & 63;               // 0..63  (unit tiles)
    const int half = lane >> 4;               // 0 or 1
    const int l15  = lane & 15;

    const float* ap = x + (size_t)(tRow * 16 + l15) * SOM_S + 2 * half;
    const float* bp = w + (size_t)(tCol * 16 + l15) * SOM_S + 2 * half;

    v8f acc = {};
    #pragma unroll 8
    for (int k = 0; k < SOM_S; k += 4) {
        v2f a = *(const v2f*)(ap + k);
        v2f b = *(const v2f*)(bp + k);
        // 8 args: (neg_a, A, neg_b, B, c_mod, C, reuse_a, reuse_b)
        acc = __builtin_amdgcn_wmma_f32_16x16x4_f32(
            false, a, false, b, (short)0, acc, false, false);
    }

    // Epilogue. C/D layout: VGPR i holds rows M=i (lanes 0-15) and M=8+i
    // (lanes 16-31), N = lane&15.
    const float wn    = ws[SOM_B + tCol * 16 + l15];
    const int   rbase = tRow * 16 + half * 8;

    float av[8];
    #pragma unroll
    for (int i = 0; i < 8; ++i) av[i] = ws[rbase + i];

    float* orow = out + (size_t)rbase * SOM_U + tCol * 16 + l15;
    #pragma unroll
    for (int i = 0; i < 8; ++i) {
        float d2 = av[i] + wn - 2.0f * acc[i];
        d2 = d2 > 0.0f ? d2 : 0.0f;           // guard tiny negative rounding
        orow[(size_t)i * SOM_U] = __builtin_sqrtf(d2);
    }
}

// ---------------------------------------------------------------------------
// Kernel 2: per-batch argmin over 1024 distances -> (row, col) BMU.
// First-occurrence tie-break matches jnp.argmin.
// ---------------------------------------------------------------------------
__global__ __launch_bounds__(256) void som_argmin(const float* __restrict__ dist,
                                                  float* __restrict__ bmus) {
    __shared__ float sv[256];
    __shared__ int   si[256];

    const int b = blockIdx.x;
    const int t = threadIdx.x;
    const float* row = dist + (size_t)b * SOM_U;

    float bestV = row[t];
    int   bestI = t;
    #pragma unroll
    for (int i = t + 256; i < SOM_U; i += 256) {
        float v = row[i];
        if (v < bestV) { bestV = v; bestI = i; }   // i increasing: keeps first
    }
    sv[t] = bestV; si[t] = bestI;
    __syncthreads();

    #pragma unroll
    for (int off = 128; off > 0; off >>= 1) {
        if (t < off) {
            float v = sv[t + off]; int ii = si[t + off];
            if (v < sv[t] || (v == sv[t] && ii < si[t])) { sv[t] = v; si[t] = ii; }
        }
        __syncthreads();
    }

    if (t == 0) {
        int idx = si[0];
        bmus[2 * b + 0] = (float)(idx / SOM_COLS);
        bmus[2 * b + 1] = (float)(idx % SOM_COLS);
    }
}

// ---------------------------------------------------------------------------
extern "C" void kernel_launch(void* const* d_in, const int* in_sizes, int n_in,
                              void* d_out, int out_size, void* d_ws, size_t ws_size,
                              hipStream_t stream) {
    (void)in_sizes; (void)n_in; (void)out_size; (void)ws_size;

    const float* x = (const float*)d_in[0];   // [512, 256]
    const float* w = (const float*)d_in[1];   // [32, 32, 256] -> [1024, 256]
    float* out = (float*)d_out;               // [512*1024] distances ++ [512*2] bmus
    float* ws  = (float*)d_ws;                // a[512] ++ wn[1024]  (6 KB)

    som_norms <<<SOM_B + SOM_U, 32, 0, stream>>>(x, w, ws);
    som_dist  <<<256, 256, 0, stream>>>(x, w, ws, out);
    som_argmin<<<SOM_B, 256, 0, stream>>>(out, out + (size_t)SOM_B * SOM_U);
}